// SubgraphEncoder_19121194402280
// MI455X (gfx1250) — compile-verified
//
#include <hip/hip_runtime.h>

// Problem sizes (fixed by the reference)
#define N_NODES  50000
#define N_EDGES  600000
#define DIM      128     // in_neurons == hidden_neurons == 128
#define HID      128
#define OUT_DIM  64
#define N_GRAPHS 512
#define BN_EPS   1e-5f

typedef __attribute__((ext_vector_type(2))) float v2f;
typedef __attribute__((ext_vector_type(8))) float v8f;

// ---------------------------------------------------------------------------
// Utility: zero fill
// ---------------------------------------------------------------------------
__global__ void zero_kernel(float* __restrict__ p, int n) {
    int i = blockIdx.x * blockDim.x + threadIdx.x;
    if (i < n) p[i] = 0.0f;
}

// ---------------------------------------------------------------------------
// Edge scatter: agg[dst] += h[src].  One 128-thread block per edge; feature f
// per thread -> coalesced gather of h[src] (hits L2: 25.6MB << 192MB) and
// f32 atomic adds into agg at L2 atomic throughput.
// ---------------------------------------------------------------------------
__global__ void scatter_add_kernel(const int* __restrict__ src,
                                   const int* __restrict__ dst,
                                   const float* __restrict__ h,
                                   float* __restrict__ agg) {
    const int e = blockIdx.x;
    const int f = threadIdx.x;
    const int s = src[e];
    const int d = dst[e];
    atomicAdd(&agg[(size_t)d * HID + f], h[(size_t)s * HID + f]);
}

// ---------------------------------------------------------------------------
// Per-graph node counts
// ---------------------------------------------------------------------------
__global__ void counts_kernel(const int* __restrict__ batch, float* __restrict__ counts) {
    int i = blockIdx.x * blockDim.x + threadIdx.x;
    if (i < N_NODES) atomicAdd(&counts[batch[i]], 1.0f);
}

// ---------------------------------------------------------------------------
// WMMA f32 GEMM: out[M x NOUT] = act( A[M x K] (+ Agg) @ W[NOUT x K]^T + bias )
//
// One wave computes a 16x16 D tile with V_WMMA_F32_16X16X4_F32, K/4 steps.
// Fragment layout (32-bit, wave32):
//   A 16x4 : lane L holds row M = L&15; VGPR0 = K = 2*(L>>4), VGPR1 = K+1
//   B 4x16 : lane L holds col N = L&15; VGPR0 = K = 2*(L>>4), VGPR1 = K+1
//   C/D    : VGPR r, lane L -> row r + 8*(L>>4), col L&15
//
// Key optimization vs naive: B (weight) fragments for the whole K reduction
// are hoisted into registers once per wave (K/4 x v2f), and the block
// grid-strides over M tiles so the weight fetch is amortized.  Inner loop is
// then a single b64 A load (+pk_add for fused h+agg) per WMMA, fully
// unrolled so the compiler can batch-issue the loads ahead of the WMMA chain.
// ---------------------------------------------------------------------------
template<bool RELU, bool ADD_AGG, int K, int NOUT>
__global__ __launch_bounds__((NOUT / 16) * 32)
void gemm_wmma_kernel(const float* __restrict__ A,
                      const float* __restrict__ Agg,
                      const float* __restrict__ W,
                      const float* __restrict__ bias,
                      float* __restrict__ out,
                      int Mtiles) {
    constexpr int KSTEPS = K / 4;
    const int wave = threadIdx.x >> 5;
    const int lane = threadIdx.x & 31;
    const int l16  = lane & 15;
    const int hq   = lane >> 4;            // 0/1 -> K offset 2*hq
    const int n_base = wave * 16;
    const int col    = n_base + l16;

    // ---- hoist B fragments (loop-invariant over M tiles) ----
    v2f bf[KSTEPS];
    const float* __restrict__ wrow = W + (size_t)col * K + 2 * hq;  // W[n][k] row-major
#pragma unroll
    for (int s = 0; s < KSTEPS; ++s) {
        bf[s].x = wrow[4 * s];
        bf[s].y = wrow[4 * s + 1];
    }
    const float bv = bias[col];

    for (int mt = blockIdx.x; mt < Mtiles; mt += gridDim.x) {
        const int m_base = mt * 16;
        const int row    = m_base + l16;
        const float* __restrict__ arow = A + (size_t)row * K + 2 * hq;
        const float* __restrict__ grow = Agg + (size_t)row * K + 2 * hq;  // dead if !ADD_AGG

        v8f acc = {};
#pragma unroll
        for (int s = 0; s < KSTEPS; ++s) {
            v2f a;
            a.x = arow[4 * s];
            a.y = arow[4 * s + 1];
            if (ADD_AGG) {
                a.x += grow[4 * s];
                a.y += grow[4 * s + 1];
            }
            // 8 args: (neg_a, A, neg_b, B, c_mod, C, reuse_a, reuse_b)
            acc = __builtin_amdgcn_wmma_f32_16x16x4_f32(
                false, a, false, bf[s], (short)0, acc, false, false);
        }

        const int r0 = m_base + 8 * hq;
#pragma unroll
        for (int r = 0; r < 8; ++r) {
            float v = acc[r] + bv;
            if (RELU) v = fmaxf(v, 0.0f);
            out[(size_t)(r0 + r) * NOUT + col] = v;
        }
    }
}

// ---------------------------------------------------------------------------
// BatchNorm statistics: per-feature sum and sum-of-squares.
// Block b strides rows; thread f owns feature column f; register-accumulate
// then one atomic per (block, feature).
// ---------------------------------------------------------------------------
__global__ void bn_stats_kernel(const float* __restrict__ h,
                                float* __restrict__ stats, int M) {
    const int f = threadIdx.x;  // 0..127
    float s = 0.0f, ss = 0.0f;
    for (int r = blockIdx.x; r < M; r += gridDim.x) {
        float v = h[(size_t)r * HID + f];
        s += v;
        ss += v * v;
    }
    atomicAdd(&stats[f], s);
    atomicAdd(&stats[HID + f], ss);
}

// ---------------------------------------------------------------------------
// BN apply (training-mode, biased var) + ReLU, in place; fused mean-pool
// accumulation into pool[graph][feature].
// ---------------------------------------------------------------------------
__global__ void bn_apply_pool_kernel(float* __restrict__ h,
                                     const float* __restrict__ stats,
                                     const float* __restrict__ gamma,
                                     const float* __restrict__ beta,
                                     const int* __restrict__ batch,
                                     float* __restrict__ pool, int M) {
    int idx = blockIdx.x * blockDim.x + threadIdx.x;
    if (idx >= M * HID) return;
    const int r = idx >> 7;
    const int f = idx & (HID - 1);
    const float invM = 1.0f / (float)M;
    const float mu  = stats[f] * invM;
    const float var = stats[HID + f] * invM - mu * mu;
    const float inv = rsqrtf(var + BN_EPS);
    float v = gamma[f] * (h[idx] - mu) * inv + beta[f];
    v = fmaxf(v, 0.0f);
    h[idx] = v;
    atomicAdd(&pool[(size_t)batch[r] * HID + f], v);
}

// ---------------------------------------------------------------------------
// Build h_final[G x 384]: concat(pool1,pool2,pool3) / max(count,1)
// ---------------------------------------------------------------------------
__global__ void build_hf_kernel(const float* __restrict__ p1,
                                const float* __restrict__ p2,
                                const float* __restrict__ p3,
                                const float* __restrict__ counts,
                                float* __restrict__ hf) {
    int idx = blockIdx.x * blockDim.x + threadIdx.x;
    if (idx >= N_GRAPHS * 3 * HID) return;
    const int g = idx / (3 * HID);
    const int k = idx - g * (3 * HID);
    const float* p = (k < HID) ? p1 : (k < 2 * HID ? p2 : p3);
    const int f = k & (HID - 1);
    const float c = fmaxf(counts[g], 1.0f);
    hf[idx] = p[(size_t)g * HID + f] / c;
}

// ---------------------------------------------------------------------------
// Row-wise L2 normalize of out[G x 64] (norm clamped to 1e-12)
// ---------------------------------------------------------------------------
__global__ void l2norm_kernel(float* __restrict__ out) {
    __shared__ float s[OUT_DIM];
    const int g = blockIdx.x;
    const int t = threadIdx.x;
    float v = out[(size_t)g * OUT_DIM + t];
    s[t] = v * v;
    __syncthreads();
    for (int off = OUT_DIM / 2; off > 0; off >>= 1) {
        if (t < off) s[t] += s[t + off];
        __syncthreads();
    }
    const float nrm = fmaxf(sqrtf(s[0]), 1e-12f);
    out[(size_t)g * OUT_DIM + t] = v / nrm;
}

// ---------------------------------------------------------------------------
// Host-side orchestration
// ---------------------------------------------------------------------------
extern "C" void kernel_launch(void* const* d_in, const int* in_sizes, int n_in,
                              void* d_out, int out_size, void* d_ws, size_t ws_size,
                              hipStream_t stream) {
    const float* x     = (const float*)d_in[0];
    const int*   ei    = (const int*)d_in[1];
    const int*   src   = ei;
    const int*   dst   = ei + N_EDGES;
    const int*   batch = (const int*)d_in[2];

    const float* W11 = (const float*)d_in[3];
    const float* b11 = (const float*)d_in[4];
    const float* W12 = (const float*)d_in[5];
    const float* b12 = (const float*)d_in[6];
    const float* g1  = (const float*)d_in[7];
    const float* be1 = (const float*)d_in[8];
    const float* W21 = (const float*)d_in[9];
    const float* b21 = (const float*)d_in[10];
    const float* W22 = (const float*)d_in[11];
    const float* b22 = (const float*)d_in[12];
    const float* g2  = (const float*)d_in[13];
    const float* be2 = (const float*)d_in[14];
    const float* W31 = (const float*)d_in[15];
    const float* b31 = (const float*)d_in[16];
    const float* W32 = (const float*)d_in[17];
    const float* b32 = (const float*)d_in[18];
    const float* g3  = (const float*)d_in[19];
    const float* be3 = (const float*)d_in[20];
    const float* linW = (const float*)d_in[21];
    const float* linb = (const float*)d_in[22];

    // ---- workspace layout (floats) ----
    const size_t NH = (size_t)N_NODES * HID;  // 6.4M floats
    float* ws    = (float*)d_ws;
    float* B0    = ws;                        // scratch slab 0
    float* B1    = B0 + NH;                   // scratch slab 1
    float* B2    = B1 + NH;                   // scratch slab 2
    float* pool1 = B2 + NH;
    float* pool2 = pool1 + (size_t)N_GRAPHS * HID;
    float* pool3 = pool2 + (size_t)N_GRAPHS * HID;
    float* counts = pool3 + (size_t)N_GRAPHS * HID;
    float* hf    = counts + N_GRAPHS;         // [G x 3H]
    float* stats = hf + (size_t)N_GRAPHS * 3 * HID;  // 2*HID

    auto zero = [&](float* p, size_t n) {
        zero_kernel<<<((int)n + 255) / 256, 256, 0, stream>>>(p, (int)n);
    };

    // pools + counts (contiguous) once per call
    zero(pool1, (size_t)3 * N_GRAPHS * HID + N_GRAPHS);
    counts_kernel<<<(N_NODES + 255) / 256, 256, 0, stream>>>(batch, counts);

    const int MTILES     = N_NODES / 16;     // 3125 M-tiles (exact)
    const int GEMM_GRID  = 512;              // grid-stride: ~6 M-tiles per block,
                                             // amortizes the per-wave B-register preload
    const int APPLY_BLKS = (N_NODES * HID + 255) / 256;

    // ---------------- one GIN layer ----------------
    auto layer = [&](const float* h_in, float* agg, float* t1, float* h_out,
                     const float* Wa, const float* ba,
                     const float* Wb, const float* bb,
                     const float* gam, const float* bet, float* pool) {
        zero(agg, NH);
        scatter_add_kernel<<<N_EDGES, HID, 0, stream>>>(src, dst, h_in, agg);
        // t1 = relu((h+agg) @ Wa^T + ba)
        gemm_wmma_kernel<true, true, HID, HID><<<GEMM_GRID, 256, 0, stream>>>(
            h_in, agg, Wa, ba, t1, MTILES);
        // h_out = t1 @ Wb^T + bb   (agg slab is dead now; h_out may alias it)
        gemm_wmma_kernel<false, false, HID, HID><<<GEMM_GRID, 256, 0, stream>>>(
            t1, nullptr, Wb, bb, h_out, MTILES);
        // BatchNorm stats + apply(ReLU) + pooled accumulation
        zero(stats, 2 * HID);
        bn_stats_kernel<<<512, HID, 0, stream>>>(h_out, stats, N_NODES);
        bn_apply_pool_kernel<<<APPLY_BLKS, 256, 0, stream>>>(
            h_out, stats, gam, bet, batch, pool, N_NODES);
    };

    // Layer 1: h_in = x,  agg = B0, t1 = B1, h1 -> B0
    layer(x,  B0, B1, B0, W11, b11, W12, b12, g1, be1, pool1);
    // Layer 2: h_in = B0, agg = B2, t1 = B1, h2 -> B2
    layer(B0, B2, B1, B2, W21, b21, W22, b22, g2, be2, pool2);
    // Layer 3: h_in = B2, agg = B0, t1 = B1, h3 -> B0
    layer(B2, B0, B1, B0, W31, b31, W32, b32, g3, be3, pool3);

    // h_final = concat(mean pools) -> [512 x 384]
    build_hf_kernel<<<(N_GRAPHS * 3 * HID + 255) / 256, 256, 0, stream>>>(
        pool1, pool2, pool3, counts, hf);

    // out = hf @ linW^T + linb  -> [512 x 64], then L2 normalize rows
    float* outp = (float*)d_out;
    gemm_wmma_kernel<false, false, 3 * HID, OUT_DIM><<<N_GRAPHS / 16, 128, 0, stream>>>(
        hf, nullptr, linW, linb, outp, N_GRAPHS / 16);
    l2norm_kernel<<<N_GRAPHS, OUT_DIM, 0, stream>>>(outp);
}